// FusionHead_46188078301471
// MI455X (gfx1250) — compile-verified
//
#include <hip/hip_runtime.h>
#include <hip/hip_bf16.h>

// CDNA5 (gfx1250) fused kernel for the FusionHead reference:
//   bu   = mean over 4 pyramid levels of point-gathered [B=64, C=26] features
//   x    = [bu | out_pc | out_context]            -> [64, 78]
//   h    = relu(x @ W1 + b1)                      -> [64, 39]
//   out  = h @ W2 + b2                            -> [64, 26]
//
// Strategy: single workgroup (384 threads = 12 wave32), everything in LDS,
// both GEMMs done with exact-fp32 V_WMMA_F32_16X16X4_F32 tiles.

typedef __attribute__((ext_vector_type(2))) float v2f;
typedef __attribute__((ext_vector_type(8))) float v8f;

#define B_    64
#define C_    26          // NB_EMO
#define KX    80          // padded 3*26 = 78 -> 80
#define N1    48          // padded 39 -> 48
#define K2    40          // padded 39 -> 40
#define N2    32          // padded 26 -> 32
#define XS    81          // LDS stride for x  (odd mod 64 -> bank-conflict free)
#define HS    49          // LDS stride for h
#define NTHREADS 384      // 12 waves: 12 L1 tiles (4x3), 8 L2 tiles (4x2)

__global__ __launch_bounds__(NTHREADS)
void FusionHead_46188078301471_kernel(
    const float* __restrict__ bu0, const int* __restrict__ cen0,
    const float* __restrict__ bu1, const int* __restrict__ cen1,
    const float* __restrict__ bu2, const int* __restrict__ cen2,
    const float* __restrict__ bu3, const int* __restrict__ cen3,
    const float* __restrict__ pc,  const float* __restrict__ ctx,
    const float* __restrict__ W1,  const float* __restrict__ b1,
    const float* __restrict__ W2,  const float* __restrict__ b2,
    float* __restrict__ out)
{
    __shared__ float xs[B_ * XS];     // x  [64][81] (cols 78,79 zero)
    __shared__ float w1s[KX * N1];    // W1 [80][48] zero-padded
    __shared__ float hs[B_ * HS];     // h  [64][49] (cols 39..47 zero)
    __shared__ float w2s[K2 * N2];    // W2 [40][32] zero-padded

    const int tid  = threadIdx.x;
    const int lane = tid & 31;
    const int wave = tid >> 5;
    const int lm   = lane & 15;       // N / M index within half-wave
    const int lh   = lane >> 4;       // half-wave select (K split / M+8 split)

    const float* bu[4]  = { bu0, bu1, bu2, bu3 };
    const int*   cen[4] = { cen0, cen1, cen2, cen3 };
    const int    res[4] = { 128, 64, 32, 16 };

    // ---- Stage 0a: gather-pool + concat -> xs ---------------------------
    for (int idx = tid; idx < B_ * KX; idx += NTHREADS) {
        const int b = idx / KX;
        const int j = idx - b * KX;
        float v = 0.0f;
        if (j < C_) {
            float s = 0.0f;
            #pragma unroll
            for (int l = 0; l < 4; ++l) {
                const int r = res[l];
                const float* p = bu[l] + (size_t)(b * C_ + j) * r * r;
                const int*   c = cen[l] + b * 8 * 2;
                #pragma unroll
                for (int ss = 0; ss < 8; ++ss) {
                    const int cx = c[2 * ss + 0];
                    const int cy = c[2 * ss + 1];
                    s += p[cy * r + cx];
                }
            }
            v = s * 0.03125f;                       // /8 subjects /4 levels
        } else if (j < 2 * C_) {
            v = pc[b * C_ + (j - C_)];
        } else if (j < 3 * C_) {
            v = ctx[b * C_ + (j - 2 * C_)];
        }                                            // j = 78,79 -> 0 (K pad)
        xs[b * XS + j] = v;
    }

    // ---- Stage 0b: zero-padded weights into LDS -------------------------
    for (int idx = tid; idx < KX * N1; idx += NTHREADS) {
        const int k = idx / N1, n = idx - k * N1;
        w1s[idx] = (k < 78 && n < 39) ? W1[k * 39 + n] : 0.0f;
    }
    for (int idx = tid; idx < K2 * N2; idx += NTHREADS) {
        const int k = idx >> 5, n = idx & 31;
        w2s[idx] = (k < 39 && n < 26) ? W2[k * C_ + n] : 0.0f;
    }
    __syncthreads();

    // ---- Stage 1: h = relu(x @ W1 + b1), fp32 WMMA 16x16x4 --------------
    {
        const int mt = wave / 3;                    // 4 M-tiles
        const int nt = wave - mt * 3;               // 3 N-tiles (48 cols)
        v8f acc = {};
        #pragma unroll
        for (int kk = 0; kk < KX / 4; ++kk) {
            const int k0 = kk * 4 + 2 * lh;         // lanes 16..31 carry K+2,K+3
            v2f a, bb;
            a.x  = xs[(mt * 16 + lm) * XS + k0];
            a.y  = xs[(mt * 16 + lm) * XS + k0 + 1];
            bb.x = w1s[(k0    ) * N1 + nt * 16 + lm];
            bb.y = w1s[(k0 + 1) * N1 + nt * 16 + lm];
            acc = __builtin_amdgcn_wmma_f32_16x16x4_f32(
                false, a, false, bb, (short)0, acc, false, false);
        }
        const int n = nt * 16 + lm;
        const float bias = (n < 39) ? b1[n] : 0.0f; // pad cols stay exactly 0
        #pragma unroll
        for (int i = 0; i < 8; ++i) {
            const float v = acc[i] + bias;          // C/D: VGPR i -> row i (+8 hi)
            hs[(mt * 16 + i + 8 * lh) * HS + n] = fmaxf(v, 0.0f);
        }
        if (lane == 0 && wave == 0) {               // zero K-pad col 39 of hs? no:
            // col 39..47 already written as relu(0+0)=0 by nt==2 tile; nothing to do
        }
    }
    __syncthreads();

    // ---- Stage 2: out = h @ W2 + b2 -------------------------------------
    if (wave < 8) {                                 // wave-uniform: EXEC stays full
        const int mt = wave >> 1;                   // 4 M-tiles
        const int nt = wave & 1;                    // 2 N-tiles (32 cols)
        v8f acc = {};
        #pragma unroll
        for (int kk = 0; kk < K2 / 4; ++kk) {
            const int k0 = kk * 4 + 2 * lh;
            v2f a, bb;
            a.x  = hs[(mt * 16 + lm) * HS + k0];
            a.y  = hs[(mt * 16 + lm) * HS + k0 + 1];
            bb.x = w2s[(k0    ) * N2 + nt * 16 + lm];
            bb.y = w2s[(k0 + 1) * N2 + nt * 16 + lm];
            acc = __builtin_amdgcn_wmma_f32_16x16x4_f32(
                false, a, false, bb, (short)0, acc, false, false);
        }
        const int n = nt * 16 + lm;
        if (n < C_) {                               // divergence only around stores
            const float bias = b2[n];
            #pragma unroll
            for (int i = 0; i < 8; ++i) {
                out[(mt * 16 + i + 8 * lh) * C_ + n] = acc[i] + bias;
            }
        }
    }
}

extern "C" void kernel_launch(void* const* d_in, const int* in_sizes, int n_in,
                              void* d_out, int out_size, void* d_ws, size_t ws_size,
                              hipStream_t stream) {
    (void)in_sizes; (void)n_in; (void)d_ws; (void)ws_size; (void)out_size;
    const float* bu0 = (const float*)d_in[0];
    const int*   c0  = (const int*)  d_in[1];
    const float* bu1 = (const float*)d_in[2];
    const int*   c1  = (const int*)  d_in[3];
    const float* bu2 = (const float*)d_in[4];
    const int*   c2  = (const int*)  d_in[5];
    const float* bu3 = (const float*)d_in[6];
    const int*   c3  = (const int*)  d_in[7];
    const float* pc  = (const float*)d_in[8];
    const float* ctx = (const float*)d_in[9];
    const float* W1  = (const float*)d_in[10];
    const float* b1  = (const float*)d_in[11];
    const float* W2  = (const float*)d_in[12];
    const float* b2  = (const float*)d_in[13];
    float* out = (float*)d_out;

    FusionHead_46188078301471_kernel<<<1, NTHREADS, 0, stream>>>(
        bu0, c0, bu1, c1, bu2, c2, bu3, c3, pc, ctx, W1, b1, W2, b2, out);
}